// Matern52withGradients_3650722201699
// MI455X (gfx1250) — compile-verified
//
#include <hip/hip_runtime.h>
#include <math.h>

typedef __attribute__((ext_vector_type(4))) float v4f;

#define TPB    256          // 8 wave32 waves per block
#define JV     4            // j-values per thread (one b128 store)
#define TILE_J (TPB * JV)   // 1024 columns per block, 4 KB x2 tile in LDS
#define ROWS   4            // rows per block (amortize LDS staging)

#define FIX_N  8192
#define FIX_M  8192

// ---------------------------------------------------------------------------
// Shared per-(i,j) math: Matern-5/2 kernel value + derivative features.
// ---------------------------------------------------------------------------
struct MPoint { float kv, dtt, dkg, d2k; };

__device__ __forceinline__ MPoint matern_point(float d, float c0, float c1,
                                               float sig2) {
  // sqrt(max(d*d, 1e-30)) == max(|d|, 1e-15)
  const float r    = fmaxf(fabsf(d), 1e-15f) * c0;
  const float er   = __expf(-r);
  const float r1   = r + 1.0f;
  const float quad = r * r * (1.0f / 3.0f);
  const float dkdf = d * r1 * er * (-c1);
  MPoint p;
  p.kv  = er * (quad + r1) * sig2;       // kttau
  p.dtt = -dkdf;                         // dkdttau
  p.dkg = dkdf;                          // dkdttau_g
  p.d2k = er * (r1 - 3.0f * quad) * c1;  // d2kdtdtau
  return p;
}

// ---------------------------------------------------------------------------
// Specialized hot kernel: N = M = 8192 baked in so every one of the 16 NT
// stores per thread is base-pointer + immediate offset (row stride 64 KB,
// half offset 32 KB; max imm 229376 B << 24-bit signed IOFFSET).
//   out[0       .. N*2M) = features  = [kttau | dkdttau]
//   out[N*2M .. 2*N*2M)  = dfeatures = [dkdf  | d2kdtdtau]
// ---------------------------------------------------------------------------
__global__ __launch_bounds__(TPB)
void matern_main_8192(const float* __restrict__ x1,
                      const float* __restrict__ x2,
                      const float* __restrict__ ws,
                      float* __restrict__ out) {
  constexpr size_t M_ = FIX_M;
  constexpr size_t M2 = 2 * M_;

  __shared__ float sx2[TILE_J];
  const int tid = threadIdx.x;
  const int j0  = blockIdx.x * TILE_J;
  const int i0  = blockIdx.y * ROWS;

  // CDNA5 async DMA: stage this block's 4 KB x2 tile global -> LDS.
  // Per-lane VDST VGPR = LDS byte offset (low 32 bits of generic shared ptr).
  {
    unsigned lds_off = (unsigned)(size_t)(&sx2[0]) + (unsigned)tid * 16u;
    unsigned goff    = (unsigned)tid * 16u;
    const float* src = x2 + j0;
    asm volatile("global_load_async_to_lds_b128 %0, %1, %2\n\t"
                 "s_wait_asynccnt 0"
                 :: "v"(lds_off), "v"(goff), "s"(src)
                 : "memory");
  }
  __syncthreads();

  const float mean = ws[0];
  const float c0   = ws[1];   // sqrt(5)/l
  const float c1   = ws[2];   // 5*sig2/(3*l^2)
  const float sig2 = ws[3];

  // One b128 load for the 4 uniform row values of this block.
  const v4f x1v = *(const v4f*)(x1 + i0);

  const v4f xv = ((const v4f*)sx2)[tid];
  float x2c[JV];
#pragma unroll
  for (int k = 0; k < JV; ++k) x2c[k] = xv[k] - mean;

  const size_t jbase = (size_t)j0 + (size_t)tid * JV;
  float* __restrict__ pF = out + (size_t)i0 * M2 + jbase;  // features rows
  float* __restrict__ pG = pF + (size_t)FIX_N * M2;        // dfeaturesdt rows

#pragma unroll
  for (int rr = 0; rr < ROWS; ++rr) {
    const float x1c = x1v[rr] - mean;
    v4f kv, dtt, dkg, d2k;
#pragma unroll
    for (int k = 0; k < JV; ++k) {
      const MPoint p = matern_point(x1c - x2c[k], c0, c1, sig2);
      kv[k] = p.kv; dtt[k] = p.dtt; dkg[k] = p.dkg; d2k[k] = p.d2k;
    }

    // 1 GiB streaming output >> 192 MB L2: non-temporal b128 stores,
    // all at compile-time immediate offsets from pF/pG.
    __builtin_nontemporal_store(kv,  (v4f*)(pF + rr * M2));
    __builtin_nontemporal_store(dtt, (v4f*)(pF + rr * M2 + M_));
    __builtin_nontemporal_store(dkg, (v4f*)(pG + rr * M2));
    __builtin_nontemporal_store(d2k, (v4f*)(pG + rr * M2 + M_));
  }
}

// ---------------------------------------------------------------------------
// Generic runtime-size fallback (only launched if sizes differ from 8192).
// ---------------------------------------------------------------------------
__global__ __launch_bounds__(TPB)
void matern_main_gen(const float* __restrict__ x1,
                     const float* __restrict__ x2,
                     const float* __restrict__ ws,
                     float* __restrict__ out,
                     int N_, int M_) {
  __shared__ float sx2[TILE_J];
  const int tid = threadIdx.x;
  const int j0  = blockIdx.x * TILE_J;
  const int i0  = blockIdx.y * ROWS;

  {
    unsigned lds_off = (unsigned)(size_t)(&sx2[0]) + (unsigned)tid * 16u;
    unsigned goff    = (unsigned)tid * 16u;
    const float* src = x2 + j0;
    asm volatile("global_load_async_to_lds_b128 %0, %1, %2\n\t"
                 "s_wait_asynccnt 0"
                 :: "v"(lds_off), "v"(goff), "s"(src)
                 : "memory");
  }
  __syncthreads();

  const float mean = ws[0], c0 = ws[1], c1 = ws[2], sig2 = ws[3];
  const v4f xv = ((const v4f*)sx2)[tid];
  float x2c[JV];
#pragma unroll
  for (int k = 0; k < JV; ++k) x2c[k] = xv[k] - mean;

  const size_t M2    = 2 * (size_t)M_;
  const size_t jbase = (size_t)j0 + (size_t)tid * JV;
  float* __restrict__ pF = out + (size_t)i0 * M2 + jbase;
  float* __restrict__ pG = pF + (size_t)N_ * M2;

#pragma unroll
  for (int rr = 0; rr < ROWS; ++rr) {
    const float x1c = x1[i0 + rr] - mean;
    v4f kv, dtt, dkg, d2k;
#pragma unroll
    for (int k = 0; k < JV; ++k) {
      const MPoint p = matern_point(x1c - x2c[k], c0, c1, sig2);
      kv[k] = p.kv; dtt[k] = p.dtt; dkg[k] = p.dkg; d2k[k] = p.d2k;
    }

    const size_t ro = (size_t)rr * M2;
    __builtin_nontemporal_store(kv,  (v4f*)(pF + ro));
    __builtin_nontemporal_store(dtt, (v4f*)(pF + ro + (size_t)M_));
    __builtin_nontemporal_store(dkg, (v4f*)(pG + ro));
    __builtin_nontemporal_store(d2k, (v4f*)(pG + ro + (size_t)M_));
  }
}

// ---------------------------------------------------------------------------
// Prep: mean(x1) + softplus-derived coefficients -> ws[0..3]
//   ws[0] = mean, ws[1] = sqrt(5)/l, ws[2] = 5*sig2/(3*l^2), ws[3] = sig2
// ---------------------------------------------------------------------------
__global__ __launch_bounds__(TPB)
void matern_prep(const float* __restrict__ x1,
                 const float* __restrict__ rawlen,
                 const float* __restrict__ rawsigf,
                 float* __restrict__ ws, int n) {
  __shared__ float red[TPB];
  const int tid = threadIdx.x;
  float s = 0.0f;
  for (int k = tid; k < n; k += TPB) s += x1[k];
  red[tid] = s;
  __syncthreads();
  for (int off = TPB / 2; off > 0; off >>= 1) {
    if (tid < off) red[tid] += red[tid + off];
    __syncthreads();
  }
  if (tid == 0) {
    const float mean = red[0] / (float)n;
    const float l    = log1pf(expf(rawlen[0]));    // softplus
    const float sigf = log1pf(expf(rawsigf[0]));   // softplus
    const float sig2 = sigf * sigf;
    ws[0] = mean;
    ws[1] = 2.23606797749978969f / l;              // sqrt(5)/l
    ws[2] = 5.0f * sig2 / (3.0f * l * l);          // c1
    ws[3] = sig2;
  }
}

// ---------------------------------------------------------------------------
extern "C" void kernel_launch(void* const* d_in, const int* in_sizes, int n_in,
                              void* d_out, int out_size, void* d_ws, size_t ws_size,
                              hipStream_t stream) {
  const float* x1      = (const float*)d_in[0];
  const float* x2      = (const float*)d_in[1];
  const float* rawlen  = (const float*)d_in[2];
  const float* rawsigf = (const float*)d_in[3];
  float*       out     = (float*)d_out;
  float*       ws      = (float*)d_ws;

  const int N_ = in_sizes[0];   // 8192
  const int M_ = in_sizes[1];   // 8192

  matern_prep<<<1, TPB, 0, stream>>>(x1, rawlen, rawsigf, ws, N_);

  dim3 grid((M_ + TILE_J - 1) / TILE_J, (N_ + ROWS - 1) / ROWS);
  if (N_ == FIX_N && M_ == FIX_M) {
    matern_main_8192<<<grid, TPB, 0, stream>>>(x1, x2, ws, out);
  } else {
    matern_main_gen<<<grid, TPB, 0, stream>>>(x1, x2, ws, out, N_, M_);
  }
}